// CayleySolitonPropagator_38259568672858
// MI455X (gfx1250) — compile-verified
//
#include <hip/hip_runtime.h>
#include <math.h>

typedef __attribute__((ext_vector_type(2))) float v2f;
typedef __attribute__((ext_vector_type(4))) float v4f;
typedef __attribute__((ext_vector_type(8))) float v8f;

#define DIMD     512
#define STRIDE   516          // 512 + 4 pad -> conflict-free B fragments
#define NSYS     16           // systems per workgroup (WMMA N dimension)
#define NTHR     256          // 8 waves
#define CG_ITERS 20
#define HALF_DT_F 0.05f
#define PI_F 3.14159265358979323846f

// ---------------- WMMA f32 16x16x4 wrapper ----------------
static __device__ __forceinline__ v8f wmma_f32_16x16x4(v2f a, v2f b, v8f c) {
  return __builtin_amdgcn_wmma_f32_16x16x4_f32(false, a, false, b, (short)0, c,
                                               false, false);
}

// Hd(delta) = (dt/2) * H entry at distance delta of the circulant band.
static __device__ __forceinline__ float band_g(int delta, const float* w, float wsum) {
  float v = (delta == 0) ? (2.0f * wsum) : 0.0f;
  int a = delta < 0 ? -delta : delta;
#pragma unroll
  for (int s = 0; s < 3; ++s) {
#pragma unroll
    for (int j = 1; j <= 5; ++j) {
      if (a == (j << s)) v -= w[s * 5 + j - 1];
    }
  }
  return HALF_DT_F * v;
}

// D_r = S_r + sign*(Hd.S_i);  D_i = S_i - sign*(Hd.S_r).
// Returns this thread's partial of dot(S, D) (pAp when sign=-1); partial
// belongs to system column N = lane&15.
static __device__ __forceinline__ float apply_cayley(
    const float* Sr, const float* Si, float* Dr, float* Di,
    const v2f afrag[5][4], float sign, int wave, int lane)
{
  const int M  = lane & 15;
  const int hi = lane >> 4;
  const int N  = M;
  float dotp = 0.0f;
  for (int t8 = 0; t8 < 4; ++t8) {
    const int I = wave * 4 + t8;           // output tile row
    v8f accR = {};
    v8f accI = {};
#pragma unroll
    for (int dd = 0; dd < 5; ++dd) {
      const int J = (I + dd + 30) & 31;    // circulant neighbor tile
#pragma unroll
      for (int kk = 0; kk < 4; ++kk) {
        const int rowb = J * 16 + kk * 4 + 2 * hi;
        v2f br = *(const v2f*)(Sr + N * STRIDE + rowb);
        v2f bi = *(const v2f*)(Si + N * STRIDE + rowb);
        accR = wmma_f32_16x16x4(afrag[dd][kk], br, accR);
        accI = wmma_f32_16x16x4(afrag[dd][kk], bi, accI);
      }
    }
    const int base = N * STRIDE + I * 16 + 8 * hi;
    v4f sr0 = *(const v4f*)(Sr + base);
    v4f sr1 = *(const v4f*)(Sr + base + 4);
    v4f si0 = *(const v4f*)(Si + base);
    v4f si1 = *(const v4f*)(Si + base + 4);
    v4f dr0, dr1, di0, di1;
#pragma unroll
    for (int r = 0; r < 4; ++r) {
      dr0[r] = sr0[r] + sign * accI[r];
      dr1[r] = sr1[r] + sign * accI[r + 4];
      di0[r] = si0[r] - sign * accR[r];
      di1[r] = si1[r] - sign * accR[r + 4];
      dotp += sr0[r] * dr0[r] + sr1[r] * dr1[r] + si0[r] * di0[r] + si1[r] * di1[r];
    }
    *(v4f*)(Dr + base)     = dr0;
    *(v4f*)(Dr + base + 4) = dr1;
    *(v4f*)(Di + base)     = di0;
    *(v4f*)(Di + base + 4) = di1;
  }
  return dotp;
}

__global__ __launch_bounds__(NTHR, 2) void cayley_cg_kernel(
    const float* __restrict__ psi_r, const float* __restrict__ psi_i,
    const float* __restrict__ alpha, const float* __restrict__ ew,
    float* __restrict__ out)
{
  __shared__ float Pr[NSYS * STRIDE], Pi_[NSYS * STRIDE];   // p (also rot)
  __shared__ float Rr[NSYS * STRIDE], Ri_[NSYS * STRIDE];   // r (also b)
  __shared__ float Xr[NSYS * STRIDE], Xi_[NSYS * STRIDE];   // x
  __shared__ float Qr[NSYS * STRIDE], Qi_[NSYS * STRIDE];   // Ap
  __shared__ float sG[5 * 256];                             // 5 band tiles, 16x16
  __shared__ float sAlpha[DIMD];
  __shared__ float red0[NTHR], red1[NTHR];
  __shared__ float sInt[NSYS], sA[NSYS], sBeta[NSYS], sRz[NSYS], sBnorm[NSYS];
  __shared__ int   sDone[NSYS];

  const int t    = threadIdx.x;
  const int wave = t >> 5, lane = t & 31;
  const int sys  = t >> 4, ch   = t & 15;
  const long sysBase = (long)blockIdx.x * NSYS;

  // ---- uniform scalars ----
  float w[15];
  float wsum = 0.0f;
#pragma unroll
  for (int k = 0; k < 15; ++k) { w[k] = ew[k]; wsum += w[k]; }
  const float dk     = HALF_DT_F * 2.0f * wsum;
  const float inv_s2 = 1.0f / (1.0f + dk * dk);

  // ---- cooperative build of the 5 circulant band tiles (Hd) in LDS ----
#pragma unroll
  for (int q = 0; q < 5; ++q) {
    const int idx = t + NTHR * q;          // 0..1279
    const int dd = idx >> 8, rc = idx & 255;
    const int row = rc >> 4, col = rc & 15;
    sG[idx] = band_g(16 * (dd - 2) + col - row, w, wsum);
  }

  // ---- stage |alpha| and psi panels ----
  sAlpha[t]       = fabsf(alpha[t]);
  sAlpha[t + 256] = fabsf(alpha[t + 256]);
  const float* gr = psi_r + sysBase * DIMD;
  const float* gi = psi_i + sysBase * DIMD;
  for (int i = 0; i < 32; ++i) {
    const int idx = t + NTHR * i;
    const int s_ = idx >> 9, d_ = idx & 511;
    Pr[s_ * STRIDE + d_]  = gr[idx];
    Pi_[s_ * STRIDE + d_] = gi[idx];
  }
  __syncthreads();

  // ---- load A fragments from LDS table (kept in VGPRs; no remat chains) ----
  v2f afrag[5][4];
  {
    const int M = lane & 15, hi = lane >> 4;
#pragma unroll
    for (int dd = 0; dd < 5; ++dd) {
#pragma unroll
      for (int kk = 0; kk < 4; ++kk) {
        afrag[dd][kk] = *(const v2f*)(sG + dd * 256 + M * 16 + kk * 4 + 2 * hi);
      }
    }
  }

  // ---- pass 1: intensity sum (== norm_in) ----
  float acc = 0.0f;
  for (int i = 0; i < 32; ++i) {
    const int off = sys * STRIDE + ch + 16 * i;
    const float a_ = Pr[off], b_ = Pi_[off];
    acc += a_ * a_ + b_ * b_;
  }
  red0[t] = acc; __syncthreads();
  if (t < NSYS) { float s = 0; for (int k = 0; k < 16; ++k) s += red0[t * 16 + k]; sInt[t] = s; }
  __syncthreads();

  // ---- pass 2: nonlinear rotation + env, accumulate norm_rot ----
  {
    const float invmean = 1.0f / fmaxf(sInt[sys] * (1.0f / 512.0f), 1e-6f);
    acc = 0.0f;
    for (int i = 0; i < 32; ++i) {
      const int d = ch + 16 * i, off = sys * STRIDE + d;
      const float r0 = Pr[off], i0 = Pi_[off];
      const float In = (r0 * r0 + i0 * i0) * invmean;
      const float ph = PI_F * (1.0f - 2.0f * __expf(-In));
      float sp, cp; __sincosf(ph, &sp, &cp);
      float rr_ = r0 * cp - i0 * sp;
      float ri_ = r0 * sp + i0 * cp;
      const float env = fminf(1.0f + sAlpha[d] * In * In, 10.0f);
      rr_ /= env; ri_ /= env;
      Pr[off] = rr_; Pi_[off] = ri_;
      acc += rr_ * rr_ + ri_ * ri_;
    }
  }
  red0[t] = acc; __syncthreads();
  if (t < NSYS) {
    float s = 0; for (int k = 0; k < 16; ++k) s += red0[t * 16 + k];
    sA[t] = fminf(sqrtf((sInt[t] + 1e-8f) / (s + 1e-8f)), 10.0f);
  }
  __syncthreads();
  {
    const float sc = sA[sys];
    for (int i = 0; i < 32; ++i) {
      const int off = sys * STRIDE + ch + 16 * i;
      Pr[off] *= sc; Pi_[off] *= sc;
    }
  }
  __syncthreads();

  // ---- rhs = A_plus(rot): R <- P via WMMA ----
  (void)apply_cayley(Pr, Pi_, Rr, Ri_, afrag, +1.0f, wave, lane);
  __syncthreads();

  // ---- CG init: x=0, p=Mprec(r), bnorm, rz ----
  {
    float pb = 0.0f, prz = 0.0f;
    for (int i = 0; i < 32; ++i) {
      const int off = sys * STRIDE + ch + 16 * i;
      const float r_ = Rr[off], ii = Ri_[off];
      const float zr = (r_ - dk * ii) * inv_s2;
      const float zi = (dk * r_ + ii) * inv_s2;
      Pr[off] = zr; Pi_[off] = zi;
      Xr[off] = 0.0f; Xi_[off] = 0.0f;
      pb  += r_ * r_ + ii * ii;
      prz += r_ * zr + ii * zi;
    }
    red0[t] = pb; red1[t] = prz;
  }
  __syncthreads();
  if (t < NSYS) {
    float sb = 0, sz = 0;
    for (int k = 0; k < 16; ++k) { sb += red0[t * 16 + k]; sz += red1[t * 16 + k]; }
    sBnorm[t] = sqrtf(sb); sRz[t] = sz; sDone[t] = 0;
  }
  __syncthreads();

  // ---- CG loop (fixed 20 iterations, done-masked like reference) ----
  for (int it = 0; it < CG_ITERS; ++it) {
    // Q = A_minus(p); pAp partial comes back fused (column N = t&15)
    const float pap = apply_cayley(Pr, Pi_, Qr, Qi_, afrag, -1.0f, wave, lane);
    red0[t] = pap;
    __syncthreads();
    if (t < NSYS) {
      float s = 0;
#pragma unroll
      for (int k = 0; k < 16; ++k) s += red0[t + 16 * k];   // all partials of column t
      sA[t] = sDone[t] ? 0.0f : sRz[t] / (s + 1e-30f);
    }
    __syncthreads();

    {
      const float a_ = sA[sys];
      float p1 = 0.0f, p2 = 0.0f;
      for (int i = 0; i < 32; ++i) {
        const int off = sys * STRIDE + ch + 16 * i;
        const float pr = Pr[off], pi = Pi_[off];
        Xr[off] += a_ * pr; Xi_[off] += a_ * pi;
        const float r_ = Rr[off] - a_ * Qr[off];
        const float ii = Ri_[off] - a_ * Qi_[off];
        Rr[off] = r_; Ri_[off] = ii;
        const float zr = (r_ - dk * ii) * inv_s2;
        const float zi = (dk * r_ + ii) * inv_s2;
        p1 += r_ * zr + ii * zi;   // rz_new
        p2 += r_ * r_ + ii * ii;   // ||r||^2
      }
      red0[t] = p1; red1[t] = p2;
    }
    __syncthreads();
    if (t < NSYS) {
      float rzn = 0, rr2 = 0;
      for (int k = 0; k < 16; ++k) { rzn += red0[t * 16 + k]; rr2 += red1[t * 16 + k]; }
      const int dn = sDone[t];
      sBeta[t] = dn ? 0.0f : rzn / (sRz[t] + 1e-30f);
      if (!dn) sRz[t] = rzn;
      if (sqrtf(rr2) <= 1e-6f * sBnorm[t]) sDone[t] = 1;
    }
    __syncthreads();

    {
      const float beta = sBeta[sys];
      for (int i = 0; i < 32; ++i) {
        const int off = sys * STRIDE + ch + 16 * i;
        const float r_ = Rr[off], ii = Ri_[off];
        const float zr = (r_ - dk * ii) * inv_s2;
        const float zi = (dk * r_ + ii) * inv_s2;
        Pr[off]  = zr + beta * Pr[off];
        Pi_[off] = zi + beta * Pi_[off];
      }
    }
    __syncthreads();
  }

  // ---- write x as interleaved [.., D, 2], coalesced float2 ----
  float2* o = (float2*)out;
  for (int i = 0; i < 32; ++i) {
    const int idx = t + NTHR * i;
    const int s_ = idx >> 9, d_ = idx & 511;
    float2 v; v.x = Xr[s_ * STRIDE + d_]; v.y = Xi_[s_ * STRIDE + d_];
    o[sysBase * DIMD + idx] = v;
  }
}

extern "C" void kernel_launch(void* const* d_in, const int* in_sizes, int n_in,
                              void* d_out, int out_size, void* d_ws, size_t ws_size,
                              hipStream_t stream) {
  const float* psi_r = (const float*)d_in[0];
  const float* psi_i = (const float*)d_in[1];
  const float* alpha = (const float*)d_in[2];
  const float* ew    = (const float*)d_in[3];
  float* out = (float*)d_out;
  const int nsys_total = in_sizes[0] / DIMD;   // 16384
  const int blocks = nsys_total / NSYS;        // 1024
  hipLaunchKernelGGL(cayley_cg_kernel, dim3(blocks), dim3(NTHR), 0, stream,
                     psi_r, psi_i, alpha, ew, out);
}